// GraphConvolutionModel_37993280700520
// MI455X (gfx1250) — compile-verified
//
#include <hip/hip_runtime.h>

typedef __attribute__((ext_vector_type(2))) float v2f;
typedef __attribute__((ext_vector_type(8))) float v8f;

static constexpr int NN = 100000;   // nodes
static constexpr int NE = 3200000;  // edges
static constexpr int NF = 128;      // input features
static constexpr int H1 = 16;       // conv1 out
static constexpr int H2 = 32;       // conv2 out
static constexpr int NG = 256;      // graphs
static constexpr int NC = 10;       // classes
static constexpr int MT = NN / 16;  // 6250 row tiles (exact)
static constexpr int LROW = 132;    // padded LDS row stride (floats); 528B = 33*16

// ---------------- degree / normalization ----------------

__global__ void k_init_deg(float* __restrict__ deg) {
  int i = blockIdx.x * blockDim.x + threadIdx.x;
  if (i < NN) deg[i] = 1.0f;  // self-loop
}

__global__ void k_deg_accum(const long long* __restrict__ ei, float* __restrict__ deg) {
  int e = blockIdx.x * blockDim.x + threadIdx.x;
  if (e < NE) {
    int d = (int)ei[NE + e];  // dst row
    atomicAdd(&deg[d], 1.0f);
  }
}

__global__ void k_rsqrt(float* __restrict__ deg) {
  int i = blockIdx.x * blockDim.x + threadIdx.x;
  if (i < NN) deg[i] = rsqrtf(deg[i]);
}

// ------- GEMM1: h1 = x @ W1 (fp32 WMMA, async global->LDS A staging) --------
// Fused epilogue also writes agg1 = h1 * dinv^2 (self-loop init, non-atomic).

__global__ __launch_bounds__(128) void k_gemm1(const float* __restrict__ x,
                                               const float* __restrict__ W1,
                                               const float* __restrict__ dinv,
                                               float* __restrict__ h1,
                                               float* __restrict__ agg1) {
  __shared__ float As[4][16 * LROW];
  const int wave = threadIdx.x >> 5;
  const int lane = threadIdx.x & 31;
  const int tile = blockIdx.x * 4 + wave;
  if (tile >= MT) return;  // wave-uniform -> EXEC all-1s for WMMA

  float* as = As[wave];
  const float* xt = x + (size_t)tile * 16 * NF;

  // Async DMA each 512B row straight into LDS (no VGPR round-trip).
  // Generic __shared__ pointer truncates to the wave-relative LDS offset.
  const unsigned lds_base = (unsigned)(uintptr_t)as;
  const unsigned long long gbase = (unsigned long long)(uintptr_t)xt;
#pragma unroll
  for (int r = 0; r < 16; ++r) {
    unsigned ldsa = lds_base + (unsigned)(r * LROW * 4 + lane * 16);
    unsigned long long ga = gbase + (unsigned long long)(r * NF * 4 + lane * 16);
    asm volatile("global_load_async_to_lds_b128 %0, %1, off"
                 :: "v"(ldsa), "v"(ga) : "memory");
  }
  asm volatile("s_wait_asynccnt 0" ::: "memory");

  const int m = lane & 15;            // A row (both half-waves)
  const int n = lane & 15;            // B/D column
  const int koff = (lane >> 4) << 1;  // lanes 0-15: K0/K1, lanes 16-31: K2/K3
  v8f c = {};
#pragma unroll
  for (int k = 0; k < NF / 4; ++k) {
    const int kb = k * 4 + koff;
    v2f a, b;
    a.x = as[m * LROW + kb];
    a.y = as[m * LROW + kb + 1];
    b.x = W1[kb * H1 + n];
    b.y = W1[(kb + 1) * H1 + n];
    c = __builtin_amdgcn_wmma_f32_16x16x4_f32(false, a, false, b, (short)0, c,
                                              false, false);
  }
  const int rb = tile * 16 + ((lane < 16) ? 0 : 8);  // D: VGPR g -> rows g/g+8
#pragma unroll
  for (int g = 0; g < 8; ++g) {
    const int row = rb + g;
    const float dv = dinv[row];
    const float v = c[g];
    h1[(size_t)row * H1 + n] = v;
    agg1[(size_t)row * H1 + n] = v * dv * dv;
  }
}

// ---------------- conv1 edge scatter (16 lanes / edge) ----------------

__global__ void k_edge1(const long long* __restrict__ ei, const float* __restrict__ h1,
                        const float* __restrict__ dinv, float* __restrict__ agg1) {
  int gid = blockIdx.x * blockDim.x + threadIdx.x;
  if (gid < NE * H1) {
    int e = gid >> 4;
    int f = gid & 15;
    int s = (int)ei[e];
    int d = (int)ei[NE + e];
    float nv = dinv[s] * dinv[d];
    atomicAdd(&agg1[d * H1 + f], h1[s * H1 + f] * nv);
  }
}

// ------- GEMM2: h2 = relu(agg1 + b1) @ W2 (bias+ReLU fused into A load) -----
// Fused epilogue also writes agg2 = h2 * dinv^2 (self-loop init).

__global__ __launch_bounds__(128) void k_gemm2(const float* __restrict__ a1,
                                               const float* __restrict__ b1,
                                               const float* __restrict__ W2,
                                               const float* __restrict__ dinv,
                                               float* __restrict__ h2,
                                               float* __restrict__ agg2) {
  const int wave = threadIdx.x >> 5;
  const int lane = threadIdx.x & 31;
  const int tile = blockIdx.x * 4 + wave;
  if (tile >= MT) return;

  const int m = lane & 15;
  const int n = lane & 15;
  const int koff = (lane >> 4) << 1;
  const float* at = a1 + (size_t)tile * 16 * H1;  // 16x16 A tile (1KB, L1-hot)
  v8f c0 = {}, c1 = {};
#pragma unroll
  for (int k = 0; k < H1 / 4; ++k) {
    const int kb = k * 4 + koff;
    v2f a, bl, bh;
    a.x = fmaxf(at[m * H1 + kb] + b1[kb], 0.0f);          // bias + ReLU fused
    a.y = fmaxf(at[m * H1 + kb + 1] + b1[kb + 1], 0.0f);
    bl.x = W2[kb * H2 + n];
    bl.y = W2[(kb + 1) * H2 + n];
    bh.x = W2[kb * H2 + 16 + n];
    bh.y = W2[(kb + 1) * H2 + 16 + n];
    c0 = __builtin_amdgcn_wmma_f32_16x16x4_f32(false, a, false, bl, (short)0, c0,
                                               false, false);
    c1 = __builtin_amdgcn_wmma_f32_16x16x4_f32(false, a, false, bh, (short)0, c1,
                                               false, false);
  }
  const int rb = tile * 16 + ((lane < 16) ? 0 : 8);
#pragma unroll
  for (int g = 0; g < 8; ++g) {
    const int row = rb + g;
    const float dv = dinv[row];
    const float sc = dv * dv;
    h2[(size_t)row * H2 + n] = c0[g];
    agg2[(size_t)row * H2 + n] = c0[g] * sc;
    h2[(size_t)row * H2 + 16 + n] = c1[g];
    agg2[(size_t)row * H2 + 16 + n] = c1[g] * sc;
  }
}

// ---------------- conv2 edge scatter (32 lanes / edge) ----------------

__global__ void k_edge2(const long long* __restrict__ ei, const float* __restrict__ h2,
                        const float* __restrict__ dinv, float* __restrict__ agg2) {
  int gid = blockIdx.x * blockDim.x + threadIdx.x;
  if (gid < NE * H2) {
    int e = gid >> 5;
    int f = gid & 31;
    int s = (int)ei[e];
    int d = (int)ei[NE + e];
    float nv = dinv[s] * dinv[d];
    atomicAdd(&agg2[d * H2 + f], h2[s * H2 + f] * nv);
  }
}

// ---------------- pooling + FC ----------------

__global__ void k_zero_pool(float* __restrict__ gsum, float* __restrict__ gcnt) {
  int i = blockIdx.x * blockDim.x + threadIdx.x;
  if (i < NG * H2) gsum[i] = 0.0f;
  if (i >= NG * H2 && i < NG * H2 + NG) gcnt[i - NG * H2] = 0.0f;
}

__global__ void k_pool(const float* __restrict__ agg2, const float* __restrict__ b2,
                       const long long* __restrict__ batch, float* __restrict__ gsum,
                       float* __restrict__ gcnt) {
  int i = blockIdx.x * blockDim.x + threadIdx.x;
  if (i < NN * H2) {
    int nIdx = i >> 5;
    int f = i & 31;
    float v = agg2[i] + b2[f];
    v = v > 0.0f ? v : 0.0f;  // bias + ReLU fused into pooling pass
    int b = (int)batch[nIdx];
    atomicAdd(&gsum[b * H2 + f], v);
    if (f == 0) atomicAdd(&gcnt[b], 1.0f);
  }
}

__global__ void k_final(const float* __restrict__ gsum, const float* __restrict__ gcnt,
                        const float* __restrict__ Wfc, const float* __restrict__ bfc,
                        float* __restrict__ out) {
  int i = blockIdx.x * blockDim.x + threadIdx.x;
  if (i < NG * NC) {
    int g = i / NC;
    int c = i % NC;
    float inv = 1.0f / fmaxf(gcnt[g], 1.0f);
    float acc = bfc[c];
#pragma unroll
    for (int k = 0; k < H2; ++k)
      acc += gsum[g * H2 + k] * inv * Wfc[k * NC + c];
    out[i] = acc;
  }
}

// ---------------- host launch ----------------

extern "C" void kernel_launch(void* const* d_in, const int* in_sizes, int n_in,
                              void* d_out, int out_size, void* d_ws, size_t ws_size,
                              hipStream_t stream) {
  (void)in_sizes; (void)n_in; (void)out_size; (void)ws_size;
  const float* x       = (const float*)d_in[0];
  const long long* ei  = (const long long*)d_in[1];  // int64 [2, E]
  const long long* bat = (const long long*)d_in[2];  // int64 [N]
  const float* W1  = (const float*)d_in[3];
  const float* b1  = (const float*)d_in[4];
  const float* W2  = (const float*)d_in[5];
  const float* b2  = (const float*)d_in[6];
  const float* Wfc = (const float*)d_in[7];
  const float* bfc = (const float*)d_in[8];
  float* out = (float*)d_out;

  char* ws = (char*)d_ws;
  size_t off = 0;
  auto carve = [&](size_t bytes) {
    float* p = (float*)(ws + off);
    off = (off + bytes + 255) & ~(size_t)255;
    return p;
  };
  float* dinv = carve((size_t)NN * 4);            // 0.4 MB
  float* h1   = carve((size_t)NN * H1 * 4);       // 6.4 MB
  float* agg1 = carve((size_t)NN * H1 * 4);       // 6.4 MB
  float* h2   = carve((size_t)NN * H2 * 4);       // 12.8 MB
  float* agg2 = carve((size_t)NN * H2 * 4);       // 12.8 MB
  float* gsum = carve((size_t)NG * H2 * 4);
  float* gcnt = carve((size_t)NG * 4);

  const int T = 256;
  k_init_deg <<<(NN + T - 1) / T, T, 0, stream>>>(dinv);
  k_deg_accum<<<(NE + T - 1) / T, T, 0, stream>>>(ei, dinv);
  k_rsqrt    <<<(NN + T - 1) / T, T, 0, stream>>>(dinv);

  k_gemm1<<<(MT + 3) / 4, 128, 0, stream>>>(x, W1, dinv, h1, agg1);
  k_edge1<<<(NE * H1 + T - 1) / T, T, 0, stream>>>(ei, h1, dinv, agg1);

  k_gemm2<<<(MT + 3) / 4, 128, 0, stream>>>(agg1, b1, W2, dinv, h2, agg2);
  k_edge2<<<(NE * H2 + T - 1) / T, T, 0, stream>>>(ei, h2, dinv, agg2);

  k_zero_pool<<<(NG * H2 + NG + T - 1) / T, T, 0, stream>>>(gsum, gcnt);
  k_pool<<<(NN * H2 + T - 1) / T, T, 0, stream>>>(agg2, b2, bat, gsum, gcnt);
  k_final<<<(NG * NC + T - 1) / T, T, 0, stream>>>(gsum, gcnt, Wfc, bfc, out);
}